// AuditorySpectrogram_78984448573788
// MI455X (gfx1250) — compile-verified
//
#include <hip/hip_runtime.h>
#include <hip/hip_bf16.h>

// Problem constants (from reference)
#define NCH       129
#define T_LEN     64000
#define BATCH     8
#define NFRAMES   250        // frames at t = 0,256,...,63744
#define LFRM      256

// Segmentation for time-parallelism (warm-up trick on stable IIRs)
#define NSEG      25
#define SEG       2560       // payload per segment (NSEG*SEG == T_LEN)
#define WARM      4096       // zero-state warm-up; 0.97^4096 ~ 0, e^-32 ~ 1.3e-14
#define CHUNK     (SEG + WARM)   // max staged samples = 6656 floats = 26.6 KB LDS
#define FR_PER_SEG (SEG / LFRM)  // 10

#define ALPHA_C   0.99221794f    // exp(-1/128)
#define BETA_C    0.88249690f    // exp(-1/8)

// Builtin's parameter type per hipcc diagnostic: generic pointer to 4 x i32 vector
typedef int v4i __attribute__((__vector_size__(16)));

__global__ __launch_bounds__(128)
void AuditorySpectrogram_78984448573788_kernel(const float* __restrict__ wav,  // (BS, T)
                                               const float* __restrict__ Bm,   // (NCH, 5)
                                               const float* __restrict__ Am,   // (NCH, 5)
                                               float* __restrict__ out)        // (BS, NCH, 250)
{
    __shared__ __align__(16) float xbuf[CHUNK];

    const int b   = blockIdx.x;          // batch
    const int g   = blockIdx.y;          // time segment
    const int tid = threadIdx.x;         // 0..127 -> channel c = tid+1

    const int t0   = g * SEG;                       // payload start (mult of 256)
    const int tbeg = (t0 - WARM) > 0 ? (t0 - WARM) : 0;  // compute start (mult of 256)
    const int n    = t0 + SEG - tbeg;               // samples to process
    const int pay  = n - SEG;                       // payload offset in chunk (mult of 256)

    // ---- Stage wav[b, tbeg : t0+SEG) into LDS (async global->LDS path) ----
    {
        const float4* srcv = (const float4*)(wav + (size_t)b * T_LEN + tbeg);
        float4*       ldsv = (float4*)xbuf;
        const int     nvec = n >> 2;   // n always divisible by 4
#if defined(__HIP_DEVICE_COMPILE__) && __has_builtin(__builtin_amdgcn_global_load_async_to_lds_b128)
        for (int i = tid; i < nvec; i += 128) {
            __builtin_amdgcn_global_load_async_to_lds_b128(
                (v4i*)(srcv + i), (v4i*)(ldsv + i), /*offset=*/0, /*cpol=*/0);
        }
  #if __has_builtin(__builtin_amdgcn_s_wait_asynccnt)
        __builtin_amdgcn_s_wait_asynccnt(0);
  #else
        asm volatile("s_wait_asynccnt 0" ::: "memory");
  #endif
#else
        for (int i = tid; i < nvec; i += 128) ldsv[i] = srcv[i];
#endif
        __syncthreads();
    }

    // ---- Per-lane filter coefficients: channel c and its neighbor c-1 ----
    const int c = tid + 1;                    // 1..128 (channel 0 output is identically 0)
    const float* Bc = Bm + 5 * c;
    const float* Ac = Am + 5 * c;
    const float* Bp = Bm + 5 * (c - 1);
    const float* Ap = Am + 5 * (c - 1);

    const float cb0 = Bc[0], cb1 = Bc[1], cb2 = Bc[2], cb3 = Bc[3], cb4 = Bc[4];
    const float ca1 = -Ac[1], ca2 = -Ac[2], ca3 = -Ac[3], ca4 = -Ac[4];
    const float pb0 = Bp[0], pb1 = Bp[1], pb2 = Bp[2], pb3 = Bp[3], pb4 = Bp[4];
    const float pa1 = -Ap[1], pa2 = -Ap[2], pa3 = -Ap[3], pa4 = -Ap[4];

    // ---- States (zero init; warm-up region absorbs the transient) ----
    float z0 = 0.f, z1 = 0.f, z2 = 0.f, z3 = 0.f;   // TDF-II state, channel c
    float w0 = 0.f, w1 = 0.f, w2 = 0.f, w3 = 0.f;   // TDF-II state, channel c-1
    float sC = 0.f, sP = 0.f;                       // beta-IIR states
    float y5 = 0.f;                                 // alpha-IIR state

    const size_t obase = (size_t)b * NCH * NFRAMES + (size_t)c * NFRAMES;
    const int    fbase = tbeg >> 8;                 // frame index of chunk sample 0

    for (int i = 0; i < n; ++i) {
        const float x = xbuf[i];

        // 4th-order IIR (transposed direct-form II), channel c
        const float yc = fmaf(cb0, x, z0);
        z0 = fmaf(ca1, yc, fmaf(cb1, x, z1));
        z1 = fmaf(ca2, yc, fmaf(cb2, x, z2));
        z2 = fmaf(ca3, yc, fmaf(cb3, x, z3));
        z3 = fmaf(ca4, yc, cb4 * x);

        // same filter for channel c-1 (independent chain -> ILP)
        const float yp = fmaf(pb0, x, w0);
        w0 = fmaf(pa1, yp, fmaf(pb1, x, w1));
        w1 = fmaf(pa2, yp, fmaf(pb2, x, w2));
        w2 = fmaf(pa3, yp, fmaf(pb3, x, w3));
        w3 = fmaf(pa4, yp, pb4 * x);

        // sigmoid (FAC == 1.0): v_exp_f32 + v_rcp_f32
        const float sc = __builtin_amdgcn_rcpf(1.0f + __expf(-yc));
        const float sp = __builtin_amdgcn_rcpf(1.0f + __expf(-yp));

        // beta-IIR (hair-cell lowpass) on both channels
        sC = fmaf(BETA_C, sC, sc);
        sP = fmaf(BETA_C, sP, sp);

        // lateral inhibition across channels + ReLU, then alpha-IIR
        const float d = fmaxf(sC - sP, 0.0f);
        y5 = fmaf(ALPHA_C, y5, d);

        // subsampled frame output (tbeg multiple of 256 -> t%256==0 <=> i%256==0)
        if (((i & (LFRM - 1)) == 0) && (i >= pay)) {
            out[obase + (size_t)(fbase + (i >> 8))] = y5;
        }
    }

    // Channel 0: y4[0] == 0 for all t -> y5 == 0 -> frames are zero.
    if (tid == 0) {
        const size_t zb = (size_t)b * NCH * NFRAMES + (size_t)(t0 >> 8);
        for (int f = 0; f < FR_PER_SEG; ++f) out[zb + f] = 0.0f;
    }
}

extern "C" void kernel_launch(void* const* d_in, const int* in_sizes, int n_in,
                              void* d_out, int out_size, void* d_ws, size_t ws_size,
                              hipStream_t stream) {
    (void)in_sizes; (void)n_in; (void)d_ws; (void)ws_size; (void)out_size;
    const float* wav = (const float*)d_in[0];   // (8, 64000) f32
    const float* B   = (const float*)d_in[1];   // (129, 5)  f32
    const float* A   = (const float*)d_in[2];   // (129, 5)  f32
    float*       out = (float*)d_out;           // (8, 129, 250) f32

    dim3 grid(BATCH, NSEG);
    dim3 block(128);
    AuditorySpectrogram_78984448573788_kernel<<<grid, block, 0, stream>>>(wav, B, A, out);
}